// GCN_27693949125321
// MI455X (gfx1250) — compile-verified
//
#include <hip/hip_runtime.h>
#include <hip/hip_bf16.h>

typedef float v2f __attribute__((ext_vector_type(2)));
typedef float v8f __attribute__((ext_vector_type(8)));

#define K_DIM 128

// ---------------- degree kernels ----------------
__global__ void deg_init_kernel(float* __restrict__ deg, int n) {
    int i = blockIdx.x * blockDim.x + threadIdx.x;
    if (i < n) deg[i] = 1.0f;   // self-loop
}

__global__ void deg_count_kernel(const int* __restrict__ dst, float* __restrict__ deg, int E) {
    int e = blockIdx.x * blockDim.x + threadIdx.x;
    if (e < E) unsafeAtomicAdd(&deg[dst[e]], 1.0f);
}

__global__ void deg_rsqrt_kernel(float* __restrict__ deg, int n) {
    int i = blockIdx.x * blockDim.x + threadIdx.x;
    if (i < n) deg[i] = 1.0f / sqrtf(deg[i]);   // deg >= 1
}

// ---------------- WMMA GEMM:  C[n x OUT] = act(A[n x 128]) @ W[128 x OUT] ----------------
// One block covers a 16-row strip; each wave owns a 16x16 output tile (OUT/16 waves).
// FUSE: apply (A + bias) then ReLU while staging the A tile into LDS (layer-2 epilogue fusion).
template<int OUT, bool FUSE>
__global__ __launch_bounds__(OUT * 2)
void gemm_wmma_kernel(const float* __restrict__ A, const float* __restrict__ W,
                      const float* __restrict__ bias, float* __restrict__ C, int n) {
    constexpr int NT = OUT * 2;            // threads per block (2 per output col)
    constexpr int LSTR = K_DIM + 4;        // 132: conflict-free fragment reads
    __shared__ float sA[16 * LSTR];

    const int row0 = blockIdx.x * 16;
    const int tid  = threadIdx.x;

    // Stage 16x128 A tile (coalesced), zero-fill OOB rows, optional bias+ReLU.
    for (int i = tid; i < 16 * K_DIM; i += NT) {
        int r = i >> 7;              // K_DIM == 128
        int c = i & (K_DIM - 1);
        int row = row0 + r;
        float v = 0.0f;
        if (row < n) {
            v = A[(size_t)row * K_DIM + c];
            if constexpr (FUSE) {
                v += bias[c];
                v = v > 0.0f ? v : 0.0f;
            }
        }
        sA[r * LSTR + c] = v;
    }
    __syncthreads();

    const int lane  = tid & 31;
    const int wv    = tid >> 5;
    const int col0  = wv * 16;
    const int nn    = lane & 15;            // output column within tile / A row (M)
    const int khalf = (lane >> 4) << 1;     // lanes 0-15 -> K+0,1 ; lanes 16-31 -> K+2,3

    v8f acc = {0.f, 0.f, 0.f, 0.f, 0.f, 0.f, 0.f, 0.f};

#pragma unroll
    for (int k = 0; k < K_DIM; k += 4) {
        v2f a, b;
        // A fragment (16x4, f32): lane holds A[M=nn][k+khalf], A[M=nn][k+khalf+1]
        a.x = sA[nn * LSTR + k + khalf];
        a.y = sA[nn * LSTR + k + khalf + 1];
        // B fragment (4x16, f32): lane holds W[k+khalf][col0+nn], W[k+khalf+1][col0+nn]
        b.x = W[(size_t)(k + khalf) * OUT + col0 + nn];
        b.y = W[(size_t)(k + khalf + 1) * OUT + col0 + nn];
        acc = __builtin_amdgcn_wmma_f32_16x16x4_f32(false, a, false, b,
                                                    (short)0, acc, false, false);
    }

    // C/D layout: VGPR r -> M = r (lanes 0-15) or r+8 (lanes 16-31), N = lane&15
    const int mbase = (lane >> 4) << 3;
#pragma unroll
    for (int r = 0; r < 8; ++r) {
        int row = row0 + mbase + r;
        if (row < n) C[(size_t)row * OUT + col0 + nn] = acc[r];
    }
}

// ---------------- self-loop initializers ----------------
// agg[i][:] = t[i][:] * dinv[i]^2        (D = 128, one float4 per thread)
__global__ void self_init128_kernel(const float* __restrict__ t, const float* __restrict__ dinv,
                                    float* __restrict__ agg, int n) {
    int i = blockIdx.x * blockDim.x + threadIdx.x;     // n*32 float4 chunks
    if (i >= n * 32) return;
    int row = i >> 5;
    float d = dinv[row];
    float w = d * d;
    float4 v = ((const float4*)t)[i];
    v.x *= w; v.y *= w; v.z *= w; v.w *= w;
    ((float4*)agg)[i] = v;
}

// out[i][:] = t[i][:] * dinv[i]^2 + b2[:]   (D = 64)
__global__ void self_init_bias64_kernel(const float* __restrict__ t, const float* __restrict__ dinv,
                                        const float* __restrict__ b, float* __restrict__ out, int n) {
    int i = blockIdx.x * blockDim.x + threadIdx.x;     // n*16 float4 chunks
    if (i >= n * 16) return;
    int row = i >> 4;
    int c4  = (i & 15) * 4;
    float d = dinv[row];
    float w = d * d;
    float4 v = ((const float4*)t)[i];
    float4 o;
    o.x = v.x * w + b[c4 + 0];
    o.y = v.y * w + b[c4 + 1];
    o.z = v.z * w + b[c4 + 2];
    o.w = v.w * w + b[c4 + 3];
    ((float4*)out)[i] = o;
}

// ---------------- edge scatter-add (L2-resident gather + native f32 atomics) ----------------
// One wave per edge; lanes split the feature dim. float4 gather = one 512B read/wave.
__global__ void edge_agg128_kernel(const int* __restrict__ src, const int* __restrict__ dst,
                                   const float* __restrict__ dinv, const float* __restrict__ t,
                                   float* __restrict__ agg, int E) {
    int gid = blockIdx.x * blockDim.x + threadIdx.x;
    int e = gid >> 5;
    if (e >= E) return;
    int lane = threadIdx.x & 31;
    int s = src[e], d = dst[e];
    float w = dinv[s] * dinv[d];
    float4 v = ((const float4*)(t + (size_t)s * 128))[lane];
    float* o = agg + (size_t)d * 128 + lane * 4;
    unsafeAtomicAdd(o + 0, v.x * w);
    unsafeAtomicAdd(o + 1, v.y * w);
    unsafeAtomicAdd(o + 2, v.z * w);
    unsafeAtomicAdd(o + 3, v.w * w);
}

__global__ void edge_agg64_kernel(const int* __restrict__ src, const int* __restrict__ dst,
                                  const float* __restrict__ dinv, const float* __restrict__ t,
                                  float* __restrict__ out, int E) {
    int gid = blockIdx.x * blockDim.x + threadIdx.x;
    int e = gid >> 5;
    if (e >= E) return;
    int lane = threadIdx.x & 31;
    int s = src[e], d = dst[e];
    float w = dinv[s] * dinv[d];
    float2 v = ((const float2*)(t + (size_t)s * 64))[lane];
    float* o = out + (size_t)d * 64 + lane * 2;
    unsafeAtomicAdd(o + 0, v.x * w);
    unsafeAtomicAdd(o + 1, v.y * w);
}

// ---------------- launcher ----------------
extern "C" void kernel_launch(void* const* d_in, const int* in_sizes, int n_in,
                              void* d_out, int out_size, void* d_ws, size_t ws_size,
                              hipStream_t stream) {
    const float* x   = (const float*)d_in[0];
    const int*   ei  = (const int*)d_in[1];
    const float* W1  = (const float*)d_in[2];
    const float* b1  = (const float*)d_in[3];
    const float* W2  = (const float*)d_in[4];
    const float* b2  = (const float*)d_in[5];
    float* out = (float*)d_out;

    const int n = in_sizes[0] / K_DIM;     // 100000
    const int E = in_sizes[1] / 2;         // 1600000
    const int* srcp = ei;                  // edge_index[0]
    const int* dstp = ei + E;              // edge_index[1]

    char* ws = (char*)d_ws;
    float* dinv = (float*)ws;                                   // n floats (deg -> dinv in place)
    size_t off  = ((size_t)n * sizeof(float) + 511) & ~(size_t)511;
    float* t1   = (float*)(ws + off);                           // n*128 (reused as t2: n*64)
    float* agg1 = (float*)(ws + off + (size_t)n * K_DIM * sizeof(float)); // n*128

    // degrees -> dinv = 1/sqrt(in_deg + 1)
    deg_init_kernel <<<(n + 255) / 256, 256, 0, stream>>>(dinv, n);
    deg_count_kernel<<<(E + 255) / 256, 256, 0, stream>>>(dstp, dinv, E);
    deg_rsqrt_kernel<<<(n + 255) / 256, 256, 0, stream>>>(dinv, n);

    // ---- layer 1: t1 = x @ W1 ; agg1 = D^-1/2 (A+I) D^-1/2 t1 ----
    gemm_wmma_kernel<128, false><<<(n + 15) / 16, 256, 0, stream>>>(x, W1, nullptr, t1, n);
    self_init128_kernel<<<(n * 32 + 255) / 256, 256, 0, stream>>>(t1, dinv, agg1, n);
    edge_agg128_kernel<<<(E * 32 + 255) / 256, 256, 0, stream>>>(srcp, dstp, dinv, t1, agg1, E);

    // ---- layer 2: t2 = relu(agg1 + b1) @ W2 (bias+ReLU fused into A staging) ----
    gemm_wmma_kernel<64, true><<<(n + 15) / 16, 128, 0, stream>>>(agg1, W2, b1, t1, n);
    self_init_bias64_kernel<<<(n * 16 + 255) / 256, 256, 0, stream>>>(t1, dinv, b2, out, n);
    edge_agg64_kernel<<<(E * 32 + 255) / 256, 256, 0, stream>>>(srcp, dstp, dinv, t1, out, E);
}